// BaseEnergyFormPhysics_47175920779452
// MI455X (gfx1250) — compile-verified
//
#include <hip/hip_runtime.h>

typedef float v2f __attribute__((ext_vector_type(2)));
typedef float v8f __attribute__((ext_vector_type(8)));

struct F3 { float x, y, z; };

__device__ __forceinline__ F3 load3(const float* __restrict__ p, int n) {
    const float* q = p + 3ull * (unsigned)n;
    F3 r; r.x = q[0]; r.y = q[1]; r.z = q[2]; return r;
}
__device__ __forceinline__ F3 sub3(F3 a, F3 b) { return {a.x - b.x, a.y - b.y, a.z - b.z}; }
__device__ __forceinline__ F3 cross3(F3 a, F3 b) {
    return {a.y * b.z - a.z * b.y, a.z * b.x - a.x * b.z, a.x * b.y - a.y * b.x};
}
__device__ __forceinline__ float dot3(F3 a, F3 b) { return a.x * b.x + a.y * b.y + a.z * b.z; }

__global__ void zero_out_kernel(float* out) { out[0] = 0.0f; }

__global__ __launch_bounds__(256) void tet_energy_kernel(
    const float* __restrict__ params, const float* __restrict__ coords,
    const float* __restrict__ us, const int4* __restrict__ conns,
    float* __restrict__ out, int nElems)
{
    const float lam = params[0];
    const float mu  = params[1];
    const float rho = params[2];

    float acc = 0.0f;
    const int tid    = blockIdx.x * blockDim.x + threadIdx.x;
    const int stride = gridDim.x * blockDim.x;

    for (int e = tid; e < nElems; e += stride) {
        if (e + 4 * stride < nElems)
            __builtin_prefetch(&conns[e + 4 * stride], 0, 0);   // global_prefetch on conn stream

        const int4 c = conns[e];
        const F3 x0 = load3(coords, c.x), x1 = load3(coords, c.y);
        const F3 x2 = load3(coords, c.z), x3 = load3(coords, c.w);
        const F3 u0 = load3(us, c.x), u1 = load3(us, c.y);
        const F3 u2 = load3(us, c.z), u3 = load3(us, c.w);

        // J columns = edge vectors
        const F3 e1 = sub3(x1, x0), e2 = sub3(x2, x0), e3 = sub3(x3, x0);
        const F3 c23 = cross3(e2, e3);
        const float det = dot3(e1, c23);
        const float inv = 1.0f / det;

        // Jinv rows == physical gradients of nodes 1..3; node 0 = -(sum)
        const F3 c31 = cross3(e3, e1);
        const F3 c12 = cross3(e1, e2);
        const F3 g1 = {c23.x * inv, c23.y * inv, c23.z * inv};
        const F3 g2 = {c31.x * inv, c31.y * inv, c31.z * inv};
        const F3 g3 = {c12.x * inv, c12.y * inv, c12.z * inv};
        const F3 g0 = {-(g1.x + g2.x + g3.x), -(g1.y + g2.y + g3.y), -(g1.z + g2.z + g3.z)};

        // grad_u[i][d] = sum_a u_a[i] * g_a[d]
        const float GxX = u0.x * g0.x + u1.x * g1.x + u2.x * g2.x + u3.x * g3.x;
        const float GxY = u0.x * g0.y + u1.x * g1.y + u2.x * g2.y + u3.x * g3.y;
        const float GxZ = u0.x * g0.z + u1.x * g1.z + u2.x * g2.z + u3.x * g3.z;
        const float GyX = u0.y * g0.x + u1.y * g1.x + u2.y * g2.x + u3.y * g3.x;
        const float GyY = u0.y * g0.y + u1.y * g1.y + u2.y * g2.y + u3.y * g3.y;
        const float GyZ = u0.y * g0.z + u1.y * g1.z + u2.y * g2.z + u3.y * g3.z;
        const float GzX = u0.z * g0.x + u1.z * g1.x + u2.z * g2.x + u3.z * g3.x;
        const float GzY = u0.z * g0.y + u1.z * g1.y + u2.z * g2.y + u3.z * g3.y;
        const float GzZ = u0.z * g0.z + u1.z * g1.z + u2.z * g2.z + u3.z * g3.z;

        const float exy = 0.5f * (GxY + GyX);
        const float exz = 0.5f * (GxZ + GzX);
        const float eyz = 0.5f * (GyZ + GzY);
        const float sumsq = GxX * GxX + GyY * GyY + GzZ * GzZ
                          + 2.0f * (exy * exy + exz * exz + eyz * eyz);
        const float tr  = GxX + GyY + GzZ;
        const float uqz = 0.25f * (u0.z + u1.z + u2.z + u3.z);

        const float psi = mu * sumsq + 0.5f * lam * tr * tr - rho * uqz;
        acc += (fabsf(det) * (1.0f / 6.0f)) * psi;
    }

    // ---- wave reduction: one exact f32 WMMA (A = ones 16x4) folds 32 partials
    //      into 16 column sums (row 0 of D), then 4 xor-shuffles finish. ----
    const int lane = threadIdx.x & 31;
    float v;
#if __has_builtin(__builtin_amdgcn_wmma_f32_16x16x4_f32)
    {
        v2f a; a[0] = 1.0f; a[1] = 1.0f;      // all-ones A (K=0..3 across lane halves)
        v2f b; b[0] = acc;  b[1] = 0.0f;      // each lane's partial occupies one B slot
        v8f d = {};
        d = __builtin_amdgcn_wmma_f32_16x16x4_f32(
                /*neg_a=*/false, a, /*neg_b=*/false, b,
                /*c_mod=*/(short)0, d, /*reuse_a=*/false, /*reuse_b=*/false);
        v = d[0];                             // lanes 0..15: D[0][n]; lanes 16..31: dup row 8
        v = (lane < 16) ? v : 0.0f;
        v += __shfl_xor(v, 1, 32);
        v += __shfl_xor(v, 2, 32);
        v += __shfl_xor(v, 4, 32);
        v += __shfl_xor(v, 8, 32);
    }
#else
    v = acc;
    for (int o = 16; o; o >>= 1) v += __shfl_xor(v, o, 32);
#endif

    __shared__ float bsum;
    if (threadIdx.x == 0) bsum = 0.0f;
    __syncthreads();
    if (lane == 0) atomicAdd(&bsum, v);       // ds atomic per wave
    __syncthreads();
    if (threadIdx.x == 0) atomicAdd(out, bsum);  // one global f32 atomic per block
}

extern "C" void kernel_launch(void* const* d_in, const int* in_sizes, int n_in,
                              void* d_out, int out_size, void* d_ws, size_t ws_size,
                              hipStream_t stream) {
    const float* params = (const float*)d_in[0];
    const float* coords = (const float*)d_in[1];
    const float* us     = (const float*)d_in[2];
    // d_in[3] = t (unused by the energy)
    const int4*  conns  = (const int4*)d_in[4];
    float* out = (float*)d_out;

    const int nElems = in_sizes[4] / 4;

    zero_out_kernel<<<1, 1, 0, stream>>>(out);

    const int threads = 256;
    const int blocks  = 2048;   // ~3.8 elements per lane, 8 wave32 per block
    tet_energy_kernel<<<blocks, threads, 0, stream>>>(params, coords, us, conns, out, nElems);
}